// YOLO_Loss_26336739459719
// MI455X (gfx1250) — compile-verified
//
#include <hip/hip_runtime.h>
#include <cstdint>

// ---------------- problem constants (from reference) ----------------
#define S7      7
#define PRED_C  30      // B*5 + C
#define TGT_C   25      // 5 + C
#define NCLS    20
#define CPB     256     // cells per block
#define BLOCK   256     // threads per block (8 wave32s)

// L_COORD=5, POS_CONF=1, POS_CLS=1, L_NOOBJ=0.5, IMG=448, GRID=64, EPS=1e-6

// Async global->LDS copy helpers (CDNA5 ASYNCcnt path, GVS addressing:
// saddr = uniform 64-bit base (SGPR pair), vaddr = per-lane 32-bit byte offset,
// vdst  = per-lane LDS byte offset). Data is a read-once stream against a
// 192 MB L2, so use the non-temporal load hint.
__device__ __forceinline__ void async_cp_b128(unsigned lds_off, unsigned goff,
                                              const float* base) {
  asm volatile("global_load_async_to_lds_b128 %0, %1, %2 th:TH_LOAD_NT"
               :: "v"(lds_off), "v"(goff), "s"(base) : "memory");
}
__device__ __forceinline__ void async_cp_b32(unsigned lds_off, unsigned goff,
                                             const float* base) {
  asm volatile("global_load_async_to_lds_b32 %0, %1, %2 th:TH_LOAD_NT"
               :: "v"(lds_off), "v"(goff), "s"(base) : "memory");
}
__device__ __forceinline__ void wait_async0() {
  asm volatile("s_wait_asynccnt 0x0" ::: "memory");
}

__global__ __launch_bounds__(BLOCK) void yolo_loss_partial(
    const float* __restrict__ pred,   // [N,7,7,30]
    const float* __restrict__ tgt,    // [N,7,7,25]
    float* __restrict__ partials,     // [gridDim.x]
    int ncells)
{
  __shared__ float s_pred[CPB * PRED_C];   // 30720 B
  __shared__ float s_tgt [CPB * TGT_C];    // 25600 B
  __shared__ float s_red [BLOCK / 32];

  const int tid       = threadIdx.x;
  const int blockBase = blockIdx.x * CPB;
  int cells_here = ncells - blockBase;
  if (cells_here > CPB) cells_here = CPB;

  const float* gpred = pred + (size_t)blockBase * PRED_C;
  const float* gtgt  = tgt  + (size_t)blockBase * TGT_C;

  // addrspace(3) offset == low 32 bits of generic pointer on AMDGPU
  const unsigned lds_pred = (unsigned)(uintptr_t)(void*)s_pred;
  const unsigned lds_tgt  = (unsigned)(uintptr_t)(void*)s_tgt;

  const int pred_dw  = cells_here * PRED_C;      // dwords to stage
  const int tgt_dw   = cells_here * TGT_C;
  const int pred_c16 = pred_dw >> 2;             // full 16B chunks
  const int tgt_c16  = tgt_dw  >> 2;

  // Main staging: b128 async copies, fully coalesced. For the full-block case
  // (cells_here == 256) chunk counts are 1920 / 1600: multiples of 64, so
  // every wave iterates with full EXEC.
  for (int c = tid; c < pred_c16; c += BLOCK) {
    const unsigned go = (unsigned)c << 4;
    async_cp_b128(lds_pred + go, go, gpred);
  }
  for (int d = (pred_c16 << 2) + tid; d < pred_dw; d += BLOCK) {  // tail (none for full blocks)
    const unsigned go = (unsigned)d << 2;
    async_cp_b32(lds_pred + go, go, gpred);
  }
  for (int c = tid; c < tgt_c16; c += BLOCK) {
    const unsigned go = (unsigned)c << 4;
    async_cp_b128(lds_tgt + go, go, gtgt);
  }
  for (int d = (tgt_c16 << 2) + tid; d < tgt_dw; d += BLOCK) {
    const unsigned go = (unsigned)d << 2;
    async_cp_b32(lds_tgt + go, go, gtgt);
  }
  wait_async0();       // this wave's async transfers landed in LDS
  __syncthreads();     // all waves' transfers visible

  float loss = 0.0f;
  if (tid < cells_here) {
    const float* p = &s_pred[tid * PRED_C];  // stride 30 dwords: conflict-free
    const float* t = &s_tgt [tid * TGT_C];   // stride 25 dwords: conflict-free

    const int cell = blockBase + tid;
    const int xx = cell % S7;
    const int yy = (cell / S7) % S7;
    const float gx = (float)xx, gy = (float)yy;

    // ground-truth box + corners
    const float tx = t[0], ty = t[1], tw = t[2], th = t[3];
    const float obj = (t[4] > 0.5f) ? 1.0f : 0.0f;
    const float gcx = (gx + tx) * 64.0f, gcy = (gy + ty) * 64.0f;
    const float gw = tw * 448.0f, gh = th * 448.0f;
    const float g_x1 = gcx - 0.5f * gw, g_y1 = gcy - 0.5f * gh;
    const float g_x2 = gcx + 0.5f * gw, g_y2 = gcy + 0.5f * gh;
    const float ga = gw * gh;

    float iou[2], cf[2];
#pragma unroll
    for (int j = 0; j < 2; ++j) {
      const float* b = &p[5 * j];
      const float pcx = (gx + b[0]) * 64.0f, pcy = (gy + b[1]) * 64.0f;
      const float pw = b[2] * 448.0f, ph = b[3] * 448.0f;
      const float p_x1 = pcx - 0.5f * pw, p_y1 = pcy - 0.5f * ph;
      const float p_x2 = pcx + 0.5f * pw, p_y2 = pcy + 0.5f * ph;
      const float pa = pw * ph;
      float iw = fminf(p_x2, g_x2) - fmaxf(p_x1, g_x1); iw = fmaxf(iw, 0.0f);
      float ih = fminf(p_y2, g_y2) - fmaxf(p_y1, g_y1); ih = fmaxf(ih, 0.0f);
      const float inter = iw * ih;
      iou[j] = inter / (pa + ga - inter + 1e-6f);
      cf[j]  = b[4];
    }

    // argmax with first-index tie-break (matches jnp.argmax)
    const int r = (iou[1] > iou[0]) ? 1 : 0;
    const float ro0 = (r == 0) ? obj : 0.0f;
    const float ro1 = (r == 1) ? obj : 0.0f;

    // no-object confidence term (applies to every non-responsible slot)
    loss = 0.5f * ((1.0f - ro0) * cf[0] * cf[0] +
                   (1.0f - ro1) * cf[1] * cf[1]);

    // responsible-box terms (scaled by obj)
    const float* br = &p[5 * r];
    const float dx = br[0] - tx, dy = br[1] - ty;
    const float dxy = dx * dx + dy * dy;
    const float sw = sqrtf(fmaxf(br[2], 0.0f)) - sqrtf(fmaxf(tw, 0.0f));
    const float sh = sqrtf(fmaxf(br[3], 0.0f)) - sqrtf(fmaxf(th, 0.0f));
    const float dwh = sw * sw + sh * sh;
    const float dconf = cf[r] - iou[r];

    float cls = 0.0f;
#pragma unroll
    for (int c = 0; c < NCLS; ++c) {
      const float d = p[10 + c] - t[5 + c];
      cls += d * d;
    }
    loss += obj * (5.0f * (dxy + dwh) + dconf * dconf + cls);
  }

  // ---- deterministic block reduction (wave32 shuffles, width 32) ----
#pragma unroll
  for (int off = 16; off > 0; off >>= 1)
    loss += __shfl_down(loss, off, 32);

  const int lane = tid & 31, wid = tid >> 5;
  if (lane == 0) s_red[wid] = loss;
  __syncthreads();
  if (tid == 0) {
    float s = 0.0f;
#pragma unroll
    for (int w = 0; w < BLOCK / 32; ++w) s += s_red[w];
    partials[blockIdx.x] = s;
  }
}

__global__ __launch_bounds__(256) void yolo_loss_reduce(
    const float* __restrict__ partials, int n, float scale,
    float* __restrict__ out)
{
  __shared__ float s[256];
  float v = 0.0f;
  for (int i = threadIdx.x; i < n; i += 256) v += partials[i];  // fixed order
  s[threadIdx.x] = v;
  __syncthreads();
#pragma unroll
  for (int off = 128; off > 0; off >>= 1) {
    if (threadIdx.x < off) s[threadIdx.x] += s[threadIdx.x + off];
    __syncthreads();
  }
  if (threadIdx.x == 0) out[0] = s[0] * scale;
}

extern "C" void kernel_launch(void* const* d_in, const int* in_sizes, int n_in,
                              void* d_out, int out_size, void* d_ws, size_t ws_size,
                              hipStream_t stream) {
  const float* pred = (const float*)d_in[0];
  const float* tgt  = (const float*)d_in[1];
  float* out        = (float*)d_out;
  float* partials   = (float*)d_ws;

  const int N      = in_sizes[0] / (S7 * S7 * PRED_C);   // 16384
  const int ncells = N * S7 * S7;                        // 802816
  const int nblk   = (ncells + CPB - 1) / CPB;           // 3136

  yolo_loss_partial<<<nblk, BLOCK, 0, stream>>>(pred, tgt, partials, ncells);
  yolo_loss_reduce<<<1, 256, 0, stream>>>(partials, nblk, 1.0f / (float)N, out);
}